// CPC_292057776614
// MI455X (gfx1250) — compile-verified
//
#include <hip/hip_runtime.h>
#include <math.h>

#define D_DIM   1280
#define NKCAND  17
#define STRIDE  1284        // D + 4 pad: bank = (4*row + k) % 64 -> conflict-free
#define MROWS   32          // output rows per workgroup (two 16-row WMMA M-tiles)
#define NBLK    280         // 8960 / 32
#define PTOT    8960

typedef __attribute__((ext_vector_type(2))) float v2f;
typedef __attribute__((ext_vector_type(8))) float v8f;

__global__ __launch_bounds__(256) void cpc_fused(
    const float* __restrict__ ctx, const float* __restrict__ enc,
    const float* __restrict__ Wk_w, const float* __restrict__ Wk_b,
    const int* __restrict__ ctx_idx, const int* __restrict__ cand_idx,
    float* __restrict__ partial)   // [NBLK][2] {loss_sum, correct_sum}
{
    __shared__ float lds[MROWS * STRIDE];    // phase 1: c-tile, phase 2: pred-tile
    __shared__ float ldsd[MROWS * NKCAND];   // dots

    const int tid = threadIdx.x;
    const int blk = blockIdx.x;

    // block -> (step s, p_base): 32-row chunks per step = {84,70,56,42,28}
    const int chunk_off[6] = {0, 84, 154, 196, 238, 280};
    const int p_off[6]     = {0, 2688, 4928, 6720, 8064, 8960};
    int s = 0;
    if (blk >= chunk_off[1]) s = 1;
    if (blk >= chunk_off[2]) s = 2;
    if (blk >= chunk_off[3]) s = 3;
    if (blk >= chunk_off[4]) s = 4;
    const int p_base = p_off[s] + (blk - chunk_off[s]) * MROWS;

    // ---- gather 32 context rows into LDS (float4, coalesced) ----
    {
        const int rr = tid >> 3;          // row 0..31 (8 threads per row)
        const int cb = tid & 7;
        const int ci = ctx_idx[p_base + rr];
        const float* crow = ctx + (size_t)ci * D_DIM;
#pragma unroll
        for (int i = 0; i < 40; ++i) {
            const int col = cb * 4 + i * 32;
            const float4 v = *(const float4*)(crow + col);
            *(float4*)&lds[rr * STRIDE + col] = v;
        }
    }
    __syncthreads();

    // ---- GEMM: pred = c @ W^T + b   via V_WMMA_F32_16X16X4_F32 ----
    // Each wave: 10 N-tiles x 2 M-tiles -> each B load feeds two WMMAs.
    const int wv = tid >> 5;
    const int ln = tid & 31;
    const int r  = ln & 15;
    const int hi = ln >> 4;
    const int n_base = wv * 160;          // each wave owns 160 output columns

    const float* Wrow = Wk_w + (size_t)s * D_DIM * D_DIM;
    const float* bptr[10];
#pragma unroll
    for (int j = 0; j < 10; ++j)
        bptr[j] = Wrow + (size_t)(n_base + j * 16 + r) * D_DIM + 2 * hi;

    const float* aptr0 = &lds[r * STRIDE + 2 * hi];          // rows  0..15
    const float* aptr1 = &lds[(16 + r) * STRIDE + 2 * hi];   // rows 16..31

    v8f acc0[10] = {};
    v8f acc1[10] = {};

    for (int k = 0; k < D_DIM; k += 4) {
        const v2f a0 = *(const v2f*)(aptr0 + k);   // A: 16x4 f32 per ISA layout
        const v2f a1 = *(const v2f*)(aptr1 + k);
#pragma unroll
        for (int j = 0; j < 10; ++j) {
            const v2f b2 = *(const v2f*)(bptr[j] + k);  // B: 4x16 = W^T tile
            acc0[j] = __builtin_amdgcn_wmma_f32_16x16x4_f32(
                false, a0, false, b2, (short)0, acc0[j], false, false);
            acc1[j] = __builtin_amdgcn_wmma_f32_16x16x4_f32(
                false, a1, false, b2, (short)0, acc1[j], false, false);
        }
    }

    // bias: lane's output column N = n_base + 16*j + r
    const float* bias = Wk_b + s * D_DIM;
#pragma unroll
    for (int j = 0; j < 10; ++j) {
        const float bv = bias[n_base + j * 16 + r];
#pragma unroll
        for (int i = 0; i < 8; ++i) { acc0[j][i] += bv; acc1[j][i] += bv; }
    }

    __syncthreads();   // all waves done reading the c-tile

    // store pred tiles: D layout -> element (M = i + 8*hi, N = n_base+16j+r)
#pragma unroll
    for (int j = 0; j < 10; ++j) {
        const int n = n_base + j * 16 + r;
#pragma unroll
        for (int i = 0; i < 8; ++i) {
            lds[(i + 8 * hi) * STRIDE + n]        = acc0[j][i];
            lds[(16 + i + 8 * hi) * STRIDE + n]   = acc1[j][i];
        }
    }
    __syncthreads();

    // ---- dots: 32 rows x 17 candidates, 68 dots per wave ----
    for (int t = wv; t < MROWS * NKCAND; t += 8) {
        const int p = t / NKCAND;
        const int k = t - p * NKCAND;
        const int ci = cand_idx[(p_base + p) * NKCAND + k];
        const float* cv = enc + (size_t)ci * D_DIM;
        const float* pr = &lds[p * STRIDE];
        float part = 0.0f;
        for (int d = ln; d < D_DIM; d += 32)
            part += pr[d] * cv[d];
#pragma unroll
        for (int off = 16; off > 0; off >>= 1)
            part += __shfl_xor(part, off, 32);
        if (ln == 0) ldsd[t] = part;
    }
    __syncthreads();

    // ---- log-softmax loss + argmax accuracy (wave 0, one row per lane) ----
    if (wv == 0) {
        const float* dp = &ldsd[ln * NKCAND];
        const float d0 = dp[0];
        float m = d0, mrest = -INFINITY;
#pragma unroll
        for (int k = 1; k < NKCAND; ++k) {
            const float dk = dp[k];
            m = fmaxf(m, dk);
            mrest = fmaxf(mrest, dk);
        }
        float se = 0.0f;
#pragma unroll
        for (int k = 0; k < NKCAND; ++k) se += expf(dp[k] - m);
        float lossv = logf(se) + m - d0;               // -log_softmax[0]
        float corrv = (d0 >= mrest) ? 1.0f : 0.0f;     // argmax == 0 (ties -> 0)
#pragma unroll
        for (int off = 16; off > 0; off >>= 1) {
            lossv += __shfl_xor(lossv, off, 32);
            corrv += __shfl_xor(corrv, off, 32);
        }
        if (ln == 0) {
            partial[blk * 2 + 0] = lossv;
            partial[blk * 2 + 1] = corrv;
        }
    }
}

__global__ __launch_bounds__(32) void cpc_finalize(
    const float* __restrict__ partial, float* __restrict__ out)
{
    const int ln = threadIdx.x;
    float l = 0.0f, c = 0.0f;
    for (int i = ln; i < NBLK; i += 32) {
        l += partial[2 * i + 0];
        c += partial[2 * i + 1];
    }
#pragma unroll
    for (int off = 16; off > 0; off >>= 1) {
        l += __shfl_xor(l, off, 32);
        c += __shfl_xor(c, off, 32);
    }
    if (ln == 0) {
        out[0] = l / (float)PTOT;
        out[1] = c / (float)PTOT;
    }
}

extern "C" void kernel_launch(void* const* d_in, const int* in_sizes, int n_in,
                              void* d_out, int out_size, void* d_ws, size_t ws_size,
                              hipStream_t stream) {
    const float* ctx      = (const float*)d_in[0];
    const float* enc      = (const float*)d_in[1];
    const float* Wk_w     = (const float*)d_in[2];
    const float* Wk_b     = (const float*)d_in[3];
    const int*   ctx_idx  = (const int*)d_in[4];
    const int*   cand_idx = (const int*)d_in[5];
    float* partial = (float*)d_ws;          // 280*2 floats, fully rewritten each call

    cpc_fused<<<NBLK, 256, 0, stream>>>(ctx, enc, Wk_w, Wk_b, ctx_idx, cand_idx, partial);
    cpc_finalize<<<1, 32, 0, stream>>>(partial, (float*)d_out);
}